// SWD22_28449863369566
// MI455X (gfx1250) — compile-verified
//
#include <hip/hip_runtime.h>
#include <hip/hip_bf16.h>

#define BATCH 16
#define VLEN  4096
#define DV    256
#define WIN   16

// Batcher odd-even merge sort network for 16 elements (63 compare-exchanges).
// All indices are literals -> the local array is fully SROA'd into VGPRs,
// lowering to v_min_num_f32 / v_max_num_f32 pairs.
__device__ __forceinline__ void sort16(float a[16]) {
#define CE(i, j)                                   \
  {                                                \
    float lo_ = fminf(a[i], a[j]);                 \
    float hi_ = fmaxf(a[i], a[j]);                 \
    a[i] = lo_;                                    \
    a[j] = hi_;                                    \
  }
  CE(0,1) CE(2,3) CE(4,5) CE(6,7) CE(8,9) CE(10,11) CE(12,13) CE(14,15)
  CE(0,2) CE(1,3) CE(4,6) CE(5,7) CE(8,10) CE(9,11) CE(12,14) CE(13,15)
  CE(1,2) CE(5,6) CE(9,10) CE(13,14)
  CE(0,4) CE(1,5) CE(2,6) CE(3,7) CE(8,12) CE(9,13) CE(10,14) CE(11,15)
  CE(2,4) CE(3,5) CE(10,12) CE(11,13)
  CE(1,2) CE(3,4) CE(5,6) CE(9,10) CE(11,12) CE(13,14)
  CE(0,8) CE(1,9) CE(2,10) CE(3,11) CE(4,12) CE(5,13) CE(6,14) CE(7,15)
  CE(4,8) CE(5,9) CE(6,10) CE(7,11)
  CE(2,4) CE(3,5) CE(6,8) CE(7,9) CE(10,12) CE(11,13)
  CE(1,2) CE(3,4) CE(5,6) CE(7,8) CE(9,10) CE(11,12) CE(13,14)
#undef CE
}

// One block per (batch, window-of-16-rows). 256 threads; thread t owns column d=t.
__global__ __launch_bounds__(256) void swd22_kernel(const float* __restrict__ v,
                                                    const int* __restrict__ bfly,
                                                    float* __restrict__ out) {
  __shared__ int      s_idx[WIN * DV];   // 16 KB butterfly-index tile
  __shared__ unsigned s_max[WIN];        // abs-max per row (uint bits of non-neg f32)
  __shared__ float    s_c0[WIN];         // column-0 values per row

  const int tid = threadIdx.x;                 // == d
  const int b   = blockIdx.x / (VLEN / WIN);
  const int win = blockIdx.x % (VLEN / WIN);
  const int l0  = win * WIN;

  if (tid < WIN) s_max[tid] = 0u;

  // ---- CDNA5 async copy: 16x256 int32 index tile (16 KB, contiguous) -> LDS ----
  // GVS mode: SGPR 64-bit base + per-lane 32-bit byte offset; LDS dest offset in vdst.
  {
    unsigned           lds_base = (unsigned)(uintptr_t)(&s_idx[0]);
    unsigned           g_base   = (unsigned)(l0 * DV * (int)sizeof(int));
    unsigned long long sbase    = (unsigned long long)(uintptr_t)bfly;
#pragma unroll
    for (int i = 0; i < 4; ++i) {
      unsigned f      = (unsigned)(i * 256 + tid) * 16u;  // 16-byte units
      unsigned ldsoff = lds_base + f;
      unsigned goff   = g_base + f;
      asm volatile("global_load_async_to_lds_b128 %0, %1, %2"
                   :: "v"(ldsoff), "v"(goff), "s"(sbase)
                   : "memory");
    }
    asm volatile("s_wait_asynccnt 0" ::: "memory");
  }
  __syncthreads();  // index tile + s_max init visible to all waves

  // ---- gather 16 rows of this thread's column ----
  // Scattered 4B reads; each v element is read exactly once globally (index
  // columns are permutations), so the 192MB L2 holds each line until all of
  // its sectors are consumed by other windows. Keep these regular-temporal.
  const float* vb = v + (size_t)b * (VLEN * DV);
  float vg[16];
#pragma unroll
  for (int j = 0; j < 16; ++j) {
    int r = s_idx[j * DV + tid];
    vg[j] = vb[r * DV + tid];
  }

  // column 0 of the gathered tile == c0 (bfly_idx[:,0] is arange)
  if (tid == 0) {
#pragma unroll
    for (int j = 0; j < 16; ++j) s_c0[j] = vg[j];
  }

  // abs-max over d via LDS ds_max_u32 (non-negative float bits are order-preserving)
#pragma unroll
  for (int j = 0; j < 16; ++j) {
    atomicMax(&s_max[j], __float_as_uint(fabsf(vg[j])));
  }
  __syncthreads();

  // ---- alter value per row, add, sort both arrays, subtract, store ----
  float av[16], alt[16];
#pragma unroll
  for (int j = 0; j < 16; ++j) {
    float c    = s_c0[j];
    float cr   = s_c0[j ^ 1];                       // pair-swapped neighbor
    float sgn  = (c > cr) ? 1.0f : ((c < cr) ? -1.0f : 0.0f);
    float amax = __uint_as_float(s_max[j]);
    alt[j] = sgn * amax;
    av[j]  = vg[j] + alt[j];
  }
  sort16(av);
  sort16(alt);

  // Output is write-once / never re-read: use non-temporal stores so the 67MB
  // result stream doesn't evict the gather working set from L2.
  float* ob = out + ((size_t)b * VLEN + l0) * DV + tid;
#pragma unroll
  for (int p = 0; p < 16; ++p) {
    __builtin_nontemporal_store(av[p] - alt[p], ob + (size_t)p * DV);
  }
}

extern "C" void kernel_launch(void* const* d_in, const int* in_sizes, int n_in,
                              void* d_out, int out_size, void* d_ws, size_t ws_size,
                              hipStream_t stream) {
  // inputs (setup_inputs order): 0=q (unused), 1=k (unused), 2=v, 3=bfly_idx, 4=len_sort_window
  const float* v    = (const float*)d_in[2];
  const int*   bfly = (const int*)d_in[3];
  float*       out  = (float*)d_out;
  (void)in_sizes; (void)n_in; (void)out_size; (void)d_ws; (void)ws_size;

  dim3 grid(BATCH * (VLEN / WIN));  // 4096 blocks, one per (batch, window)
  dim3 block(DV);                   // 256 threads = 8 wave32
  swd22_kernel<<<grid, block, 0, stream>>>(v, bfly, out);
}